// tNN8MNIST_19963007992238
// MI455X (gfx1250) — compile-verified
//
#include <hip/hip_runtime.h>

// ---------------------------------------------------------------------------
// tNN (t-product DCT network), 8 layers fused per 32-column batch tile in LDS.
// fp32 WMMA 16x16x4 on gfx1250 (wave32), async global->LDS tile load.
// ---------------------------------------------------------------------------

#define NSLICE 28             // tube length / slice count
#define NBV    32             // valid batch columns per workgroup
#define NBP    36             // group stride: 16B aligned, bank-safe
#define NCOLS  (NSLICE*NBP)   // 1008 active columns per LDS row
#define RSTR   1012           // LDS row stride (+4 pad floats; bank-safe)
#define NTA    63             // 1008/16 N-tiles for stages A/C
#define HALFA  32             // tile-pairing half for stages A/C
#define BATCH  65536
#define WHAT_SLICE (32*28)    // floats per W_hat slice (rows padded 28->32)
#define WHAT_LAYER (28*WHAT_SLICE)

typedef __attribute__((ext_vector_type(2))) float v2f;
typedef __attribute__((ext_vector_type(8))) float v8f;

static __device__ __forceinline__ v8f wmma4(v2f a, v2f b, v8f c) {
  // D = A(16x4) * B(4x16) + C(16x16), fp32
  return __builtin_amdgcn_wmma_f32_16x16x4_f32(
      /*neg_a=*/false, a, /*neg_b=*/false, b,
      /*c_mod=*/(short)0, c, /*reuse_a=*/false, /*reuse_b=*/false);
}

// ---------------------------------------------------------------------------
// Init kernel 1: orthonormal DCT-II matrix D (32x28, rows >=28 zero-padded)
// and its transpose DT (32x28, rows >=28 zero-padded).
// ---------------------------------------------------------------------------
__global__ void tnn_init_dct(float* __restrict__ Dg, float* __restrict__ DTg) {
  int idx = blockIdx.x * blockDim.x + threadIdx.x;
  if (idx >= 32 * 28) return;
  int r = idx / 28;
  int c = idx % 28;
  const double PI = 3.14159265358979323846;
  double s = sqrt(2.0 / 28.0);
  double dv = 0.0, dtv = 0.0;
  if (r < 28) {
    dv = s * cos(PI * (c + 0.5) * (double)r / 28.0);
    if (r == 0) dv *= 0.70710678118654752440;
    dtv = s * cos(PI * (r + 0.5) * (double)c / 28.0);
    if (c == 0) dtv *= 0.70710678118654752440;
  }
  Dg[idx]  = (float)dv;
  DTg[idx] = (float)dtv;
}

// ---------------------------------------------------------------------------
// Init kernel 2: W_hat[l][i][a][b] = sum_j D[i][j] * W_l[j][a][b]
// Stored [8][28 slices][32 rows (a, zero-padded)][28 cols (b)].
// ---------------------------------------------------------------------------
__global__ void tnn_init_what(const float* __restrict__ Dg,
                              float* __restrict__ what,
                              const float* __restrict__ W0, const float* __restrict__ W1,
                              const float* __restrict__ W2, const float* __restrict__ W3,
                              const float* __restrict__ W4, const float* __restrict__ W5,
                              const float* __restrict__ W6, const float* __restrict__ W7) {
  int idx = blockIdx.x * blockDim.x + threadIdx.x;
  if (idx >= 8 * WHAT_LAYER) return;
  int b = idx % 28;
  int t = idx / 28;
  int a = t % 32; t /= 32;
  int i = t % 28;
  int lay = t / 28;
  const float* Wp[8] = {W0, W1, W2, W3, W4, W5, W6, W7};
  int Adim = (lay == 7) ? 10 : 28;
  float v = 0.0f;
  if (a < Adim) {
    const float* W = Wp[lay];
    #pragma unroll
    for (int j = 0; j < 28; ++j)
      v += Dg[i * 28 + j] * W[(j * Adim + a) * 28 + b];
  }
  what[idx] = v;
}

// ---------------------------------------------------------------------------
// Main fused kernel. One workgroup (8 wave32) = 32 batch columns, all 8 layers.
// LDS holds two 28 x 1012 f32 buffers (ping-pong), ~227 KB.
// ---------------------------------------------------------------------------
__global__ __launch_bounds__(256, 1)
void tnn8_fused(const float* __restrict__ x,
                const float* __restrict__ Dg,
                const float* __restrict__ DTg,
                const float* __restrict__ what,
                const float* __restrict__ bb0, const float* __restrict__ bb1,
                const float* __restrict__ bb2, const float* __restrict__ bb3,
                const float* __restrict__ bb4, const float* __restrict__ bb5,
                const float* __restrict__ bb6, const float* __restrict__ bb7,
                float* __restrict__ out) {
  __shared__ float lds[2 * NSLICE * RSTR];

  const int tid    = threadIdx.x;
  const int wv     = tid >> 5;        // wave id 0..7
  const int lane   = tid & 31;
  const int lrow   = lane & 15;       // M/N position within fragment
  const int lhi    = lane >> 4;       // lane half selects K pair / row group
  const int batch0 = blockIdx.x * NBV;

  const float* biasArr[8] = {bb0, bb1, bb2, bb3, bb4, bb5, bb6, bb7};

  float* cur = lds;
  float* alt = lds + NSLICE * RSTR;

  // ---- async global->LDS load of the input tile (16B chunks, CDNA5 TDM-lite)
  // cur[j][b*NBP + n..n+3] = x[j][b][batch0 + n..n+3]
  for (int idx = tid; idx < NSLICE * NSLICE * (NBV / 4); idx += 256) {
    unsigned n    = (idx & 7) * 4;
    unsigned rest = (unsigned)idx >> 3;
    unsigned b    = rest % 28;
    unsigned j    = rest / 28;
    unsigned ldsoff = (unsigned)((j * RSTR + b * NBP + n) * 4);
    unsigned goff   = ((j * 28u + b) * BATCH + (unsigned)batch0 + n) * 4u;
    asm volatile("global_load_async_to_lds_b128 %0, %1, %2"
                 :: "v"(ldsoff), "v"(goff), "s"(x) : "memory");
  }

  // ---- preload D / DT A-fragments (overlaps with async copy) --------------
  v2f dA[2][7], dtA[2][7];
  #pragma unroll
  for (int mt = 0; mt < 2; ++mt) {
    #pragma unroll
    for (int k = 0; k < 7; ++k) {
      int r  = mt * 16 + lrow;
      int kk = k * 4 + 2 * lhi;
      dA[mt][k].x  = Dg[r * 28 + kk];
      dA[mt][k].y  = Dg[r * 28 + kk + 1];
      dtA[mt][k].x = DTg[r * 28 + kk];
      dtA[mt][k].y = DTg[r * 28 + kk + 1];
    }
  }

  // warm L2/L1 with layer-0 W_hat while the async copy drains
  for (int p = tid * 32; p < WHAT_LAYER; p += 256 * 32)
    __builtin_prefetch(what + p, 0, 1);

  asm volatile("s_wait_asynccnt 0" ::: "memory");
  __syncthreads();

  #pragma unroll 1
  for (int lay = 0; lay < 8; ++lay) {
    const int    Adim = (lay == 7) ? 10 : 28;
    const float* Wh   = what + lay * WHAT_LAYER;
    const float* bia  = biasArr[lay];

    // prefetch next layer's W_hat behind this layer's compute
    if (lay < 7) {
      const float* nw = what + (lay + 1) * WHAT_LAYER;
      for (int p = tid * 32; p < WHAT_LAYER; p += 256 * 32)
        __builtin_prefetch(nw + p, 0, 1);
    }

    // ====== Stage A: alt = D @ cur (contract slice axis), paired tiles =====
    #pragma unroll
    for (int mt = 0; mt < 2; ++mt) {
      for (int nt = wv; nt < HALFA; nt += 8) {
        const bool do1 = (nt + HALFA) < NTA;           // wave-uniform
        const int  c0  = nt * 16;
        const int  c1  = do1 ? (nt + HALFA) * 16 : c0;
        v8f acc0 = {0.f, 0.f, 0.f, 0.f, 0.f, 0.f, 0.f, 0.f};
        v8f acc1 = {0.f, 0.f, 0.f, 0.f, 0.f, 0.f, 0.f, 0.f};
        #pragma unroll
        for (int k = 0; k < 7; ++k) {
          int kr = k * 4 + 2 * lhi;
          v2f b0, b1;
          b0.x = cur[kr * RSTR + c0 + lrow];
          b0.y = cur[(kr + 1) * RSTR + c0 + lrow];
          b1.x = cur[kr * RSTR + c1 + lrow];
          b1.y = cur[(kr + 1) * RSTR + c1 + lrow];
          acc0 = wmma4(dA[mt][k], b0, acc0);
          acc1 = wmma4(dA[mt][k], b1, acc1);
        }
        #pragma unroll
        for (int r = 0; r < 8; ++r) {
          int row = mt * 16 + r + 8 * lhi;
          if (row < NSLICE) alt[row * RSTR + c0 + lrow] = acc0[r];
        }
        if (do1) {
          #pragma unroll
          for (int r = 0; r < 8; ++r) {
            int row = mt * 16 + r + 8 * lhi;
            if (row < NSLICE) alt[row * RSTR + c1 + lrow] = acc1[r];
          }
        }
      }
    }
    __syncthreads();

    // ====== Stage B: per-slice matmul, cur[i] = W_hat[i] @ alt[i] ==========
    for (int i = wv; i < NSLICE; i += 8) {
      const float* wa  = Wh + i * WHAT_SLICE;
      const float* src = alt + i * RSTR;
      float*       dst = cur + i * RSTR;
      const int mtN = (Adim == 28) ? 2 : 1;
      for (int mt = 0; mt < mtN; ++mt) {
        #pragma unroll
        for (int nt = 0; nt < 2; ++nt) {
          v8f acc = {0.f, 0.f, 0.f, 0.f, 0.f, 0.f, 0.f, 0.f};
          #pragma unroll
          for (int k = 0; k < 7; ++k) {
            int kk = k * 4 + 2 * lhi;
            v2f af, bf;
            const float* ap = wa + (mt * 16 + lrow) * 28 + kk;
            af.x = ap[0];
            af.y = ap[1];
            bf.x = src[kk * NBP + nt * 16 + lrow];
            bf.y = src[(kk + 1) * NBP + nt * 16 + lrow];
            acc = wmma4(af, bf, acc);
          }
          #pragma unroll
          for (int r = 0; r < 8; ++r) {
            int a = mt * 16 + r + 8 * lhi;
            if (a < Adim) dst[a * NBP + nt * 16 + lrow] = acc[r];
          }
        }
      }
    }
    __syncthreads();

    // ====== Stage C: alt = D^T @ cur, + bias (+ ReLU), layer 8 -> d_out ====
    const int half = (lay == 7) ? 12 : HALFA;
    const int lim  = (lay == 7) ? 23 : NTA;   // layer 8: 10*36=360 cols -> 23 tiles
    #pragma unroll
    for (int mt = 0; mt < 2; ++mt) {
      for (int nt = wv; nt < half; nt += 8) {
        const bool do1 = (nt + half) < lim;            // wave-uniform
        const int  c0  = nt * 16;
        const int  c1  = do1 ? (nt + half) * 16 : c0;
        v8f acc0 = {0.f, 0.f, 0.f, 0.f, 0.f, 0.f, 0.f, 0.f};
        v8f acc1 = {0.f, 0.f, 0.f, 0.f, 0.f, 0.f, 0.f, 0.f};
        #pragma unroll
        for (int k = 0; k < 7; ++k) {
          int kr = k * 4 + 2 * lhi;
          v2f b0, b1;
          b0.x = cur[kr * RSTR + c0 + lrow];
          b0.y = cur[(kr + 1) * RSTR + c0 + lrow];
          b1.x = cur[kr * RSTR + c1 + lrow];
          b1.y = cur[(kr + 1) * RSTR + c1 + lrow];
          acc0 = wmma4(dtA[mt][k], b0, acc0);
          acc1 = wmma4(dtA[mt][k], b1, acc1);
        }
        const int col0 = c0 + lrow, a0 = col0 / NBP, n0 = col0 - a0 * NBP;
        const int col1 = c1 + lrow, a1 = col1 / NBP, n1 = col1 - a1 * NBP;
        if (lay < 7) {
          #pragma unroll
          for (int r = 0; r < 8; ++r) {
            int row = mt * 16 + r + 8 * lhi;
            if (row < NSLICE) {
              alt[row * RSTR + col0] = fmaxf(acc0[r] + bia[row * 28 + a0], 0.0f);
            }
          }
          if (do1) {
            #pragma unroll
            for (int r = 0; r < 8; ++r) {
              int row = mt * 16 + r + 8 * lhi;
              if (row < NSLICE) {
                alt[row * RSTR + col1] = fmaxf(acc1[r] + bia[row * 28 + a1], 0.0f);
              }
            }
          }
        } else {
          if (a0 < 10 && n0 < NBV) {
            #pragma unroll
            for (int r = 0; r < 8; ++r) {
              int row = mt * 16 + r + 8 * lhi;
              if (row < NSLICE)
                out[(row * 10 + a0) * BATCH + batch0 + n0] = acc0[r] + bia[row * 10 + a0];
            }
          }
          if (do1 && a1 < 10 && n1 < NBV) {
            #pragma unroll
            for (int r = 0; r < 8; ++r) {
              int row = mt * 16 + r + 8 * lhi;
              if (row < NSLICE)
                out[(row * 10 + a1) * BATCH + batch0 + n1] = acc1[r] + bia[row * 10 + a1];
            }
          }
        }
      }
    }
    __syncthreads();

    // ping-pong buffers for next layer (next layer's input is this alt)
    float* tp = cur; cur = alt; alt = tp;
  }
}

// ---------------------------------------------------------------------------
// Launcher. Inputs: [0]=x, [1]=W1,[2]=B1, ..., [15]=W8,[16]=B8 (all f32).
// ws layout (floats): [0,896) D, [896,1792) DT, [1792, +8*28*32*28) W_hat.
// ---------------------------------------------------------------------------
extern "C" void kernel_launch(void* const* d_in, const int* in_sizes, int n_in,
                              void* d_out, int out_size, void* d_ws, size_t ws_size,
                              hipStream_t stream) {
  (void)in_sizes; (void)n_in; (void)out_size; (void)ws_size;
  const float* x = (const float*)d_in[0];
  const float* W[8];
  const float* B[8];
  for (int i = 0; i < 8; ++i) {
    W[i] = (const float*)d_in[1 + 2 * i];
    B[i] = (const float*)d_in[2 + 2 * i];
  }
  float* ws   = (float*)d_ws;
  float* Dg   = ws;
  float* DTg  = ws + 32 * 28;
  float* what = ws + 2 * 32 * 28;

  tnn_init_dct<<<(32 * 28 + 255) / 256, 256, 0, stream>>>(Dg, DTg);

  int whatElems = 8 * WHAT_LAYER;
  tnn_init_what<<<(whatElems + 255) / 256, 256, 0, stream>>>(
      Dg, what, W[0], W[1], W[2], W[3], W[4], W[5], W[6], W[7]);

  tnn8_fused<<<BATCH / NBV, 256, 0, stream>>>(
      x, Dg, DTg, what,
      B[0], B[1], B[2], B[3], B[4], B[5], B[6], B[7],
      (float*)d_out);
}